// TissueGraphNetwork_51737176047902
// MI455X (gfx1250) — compile-verified
//
#include <hip/hip_runtime.h>

#define NLAYERS  3
#define HD       64
#define NNODES   4096
#define NEDGES   131072
#define NGRAPHS  8
#define LN_EPS   1e-5f

typedef __attribute__((ext_vector_type(2))) float v2f;
typedef __attribute__((ext_vector_type(8))) float v8f;

// ---------------------------------------------------------------------------
// h = X @ W   (X: [32768 x 64] fp32, W: [64 x 64] fp32, row-major)
// fp32 WMMA: V_WMMA_F32_16X16X4_F32. One wave -> 16 rows x 64 cols strip.
//
// W is staged into LDS K-pair-interleaved:
//   Wlds[p*64 + c] = { W[2p][c], W[2p+1][c] }   (p = 0..31, c = 0..63)
// so a B fragment (two consecutive K values, one column) is a single aligned
// ds_load_b64 into an even VGPR pair -> no repacking movs before v_wmma.
//
// A 16x4 frag: lane l (M = l&15): v0 = K(k0 + (l>=16?2:0)), v1 = K(+1)
// B 4x16 frag: lane l (N = l&15): v0 = K(k0 + (l>=16?2:0)), v1 = K(+1)
// C/D 16x16:   vgpr r: lanes0-15 -> M=r, lanes16-31 -> M=r+8, N = l&15
// ---------------------------------------------------------------------------
__global__ __launch_bounds__(256)
void gemm_wmma_f32(const float* __restrict__ X, const float* __restrict__ Wg,
                   float* __restrict__ Hout)
{
    __shared__ v2f Wlds[32 * HD];             // 16 KB of the 320 KB/WGP LDS
    // stage W interleaved by K-pairs: 2048 float2 entries, 8 per thread
    #pragma unroll
    for (int i = 0; i < 8; ++i) {
        const int idx = threadIdx.x + i * 256; // 0..2047
        const int p = idx >> 6;                // K-pair index
        const int c = idx & 63;                // column
        v2f w; w.x = Wg[(2 * p) * HD + c]; w.y = Wg[(2 * p + 1) * HD + c];
        Wlds[idx] = w;
    }
    __syncthreads();

    const int wave    = threadIdx.x >> 5;
    const int lane    = threadIdx.x & 31;
    const int hl      = lane >> 4;            // lane half: 0 / 1
    const int l15     = lane & 15;
    const int rowBase = (blockIdx.x * 8 + wave) * 16;
    const int arow    = rowBase + l15;

    v8f c0 = {}, c1 = {}, c2 = {}, c3 = {};

    #pragma unroll
    for (int k0 = 0; k0 < HD; k0 += 4) {
        const int kb = k0 + hl * 2;           // this lane's first K
        const int pb = kb >> 1;               // K-pair index in LDS
        // A fragment: one 8-byte global load (two consecutive K of this row)
        const v2f a = *(const v2f*)(X + (size_t)arow * HD + kb);
        // B fragments: one ds_load_b64 each
        const v2f b0 = Wlds[pb * HD +  0 + l15];
        const v2f b1 = Wlds[pb * HD + 16 + l15];
        const v2f b2 = Wlds[pb * HD + 32 + l15];
        const v2f b3 = Wlds[pb * HD + 48 + l15];

        c0 = __builtin_amdgcn_wmma_f32_16x16x4_f32(false, a, false, b0, (short)0, c0, false, false);
        c1 = __builtin_amdgcn_wmma_f32_16x16x4_f32(false, a, false, b1, (short)0, c1, false, false);
        c2 = __builtin_amdgcn_wmma_f32_16x16x4_f32(false, a, false, b2, (short)0, c2, false, false);
        c3 = __builtin_amdgcn_wmma_f32_16x16x4_f32(false, a, false, b3, (short)0, c3, false, false);
    }

    const int srow = rowBase + hl * 8;
    #pragma unroll
    for (int r = 0; r < 8; ++r) {
        float* o = Hout + (size_t)(srow + r) * HD + l15;
        o[ 0] = c0[r];
        o[16] = c1[r];
        o[32] = c2[r];
        o[48] = c3[r];
    }
}

// ---------------------------------------------------------------------------
// Edge gather-gate-scatter: acc[dst] += h[src] * ea
// One wave per edge; lane handles 2 consecutive features (coalesced 256B/wave).
// edge_attrs is a one-shot 768 MB stream across 3 layers -> NT hint so the
// 192 MB L2 keeps the h gather table (1 MB/graph) and accumulators resident.
// ---------------------------------------------------------------------------
__global__ __launch_bounds__(256)
void edge_scatter(const float* __restrict__ Hfeat, const float* __restrict__ EA,
                  const long long* __restrict__ EI, float* __restrict__ Acc)
{
    const int  lane = threadIdx.x & 31;
    const long long eid = (long long)blockIdx.x * 8 + (threadIdx.x >> 5);
    const int  b = (int)(eid >> 17);            // / NEDGES (2^17)
    const int  e = (int)(eid & (NEDGES - 1));

    const long long* eib = EI + (size_t)b * 2 * NEDGES;
    const int src = (int)eib[e];
    const int dst = (int)eib[NEDGES + e];

    // 8-byte vector loads (lane*2 is even -> 8B aligned)
    const v2f m = *(const v2f*)(Hfeat + ((size_t)b * NNODES + src) * HD + lane * 2);
    const v2f a = __builtin_nontemporal_load(
        (const v2f*)(EA + ((size_t)b * NEDGES + e) * HD) + lane);

    float* out = Acc + ((size_t)b * NNODES + dst) * HD + lane * 2;
    unsafeAtomicAdd(out,     m.x * a.x);        // global_atomic_add_f32
    unsafeAtomicAdd(out + 1, m.y * a.y);
}

// ---------------------------------------------------------------------------
// bias + ReLU + LayerNorm(H=64) + optional residual; one wave32 per node,
// 2 features/lane, shfl_xor tree reductions (no LDS).
// ---------------------------------------------------------------------------
__global__ __launch_bounds__(256)
void post_ln(const float* __restrict__ Acc, const float* __restrict__ bias,
             const float* __restrict__ gamma, const float* __restrict__ beta,
             float* __restrict__ Xbuf, int addResidual)
{
    const int lane = threadIdx.x & 31;
    const int node = blockIdx.x * 8 + (threadIdx.x >> 5);
    const size_t base = (size_t)node * HD;

    float v0 = fmaxf(Acc[base + lane]      + bias[lane],      0.0f);
    float v1 = fmaxf(Acc[base + 32 + lane] + bias[32 + lane], 0.0f);

    float s = v0 + v1;
    #pragma unroll
    for (int m = 16; m >= 1; m >>= 1) s += __shfl_xor(s, m, 32);
    const float mean = s * (1.0f / 64.0f);

    const float d0 = v0 - mean, d1 = v1 - mean;
    float q = d0 * d0 + d1 * d1;
    #pragma unroll
    for (int m = 16; m >= 1; m >>= 1) q += __shfl_xor(q, m, 32);
    const float rstd = rsqrtf(q * (1.0f / 64.0f) + LN_EPS);

    float o0 = d0 * rstd * gamma[lane]      + beta[lane];
    float o1 = d1 * rstd * gamma[32 + lane] + beta[32 + lane];
    if (addResidual) {
        o0 += Xbuf[base + lane];
        o1 += Xbuf[base + 32 + lane];
    }
    Xbuf[base + lane]      = o0;
    Xbuf[base + 32 + lane] = o1;
}

// ---------------------------------------------------------------------------
// out = x * presence + global_emb * (1 - presence); presence from raw x0 rows.
// ---------------------------------------------------------------------------
__global__ __launch_bounds__(256)
void finalize(const float* __restrict__ X0, const float* __restrict__ Xbuf,
              const float* __restrict__ Gemb, float* __restrict__ Out)
{
    const int lane = threadIdx.x & 31;
    const int node = blockIdx.x * 8 + (threadIdx.x >> 5);   // 0..32767
    const int n    = node & (NNODES - 1);                   // node within graph
    const size_t base = (size_t)node * HD;
    const size_t gb   = (size_t)n * HD;

    float s = X0[base + lane] + X0[base + 32 + lane];
    #pragma unroll
    for (int m = 16; m >= 1; m >>= 1) s += __shfl_xor(s, m, 32);
    const float p = (s != 0.0f) ? 1.0f : 0.0f;

    Out[base + lane]      = Xbuf[base + lane]      * p + Gemb[gb + lane]      * (1.0f - p);
    Out[base + 32 + lane] = Xbuf[base + 32 + lane] * p + Gemb[gb + 32 + lane] * (1.0f - p);
}

// ---------------------------------------------------------------------------
extern "C" void kernel_launch(void* const* d_in, const int* in_sizes, int n_in,
                              void* d_out, int out_size, void* d_ws, size_t ws_size,
                              hipStream_t stream)
{
    const float*     x0    = (const float*)d_in[0];      // [8,4096,64]
    const long long* ei    = (const long long*)d_in[1];  // [8,2,131072] int64
    const float*     ea    = (const float*)d_in[2];      // [8,131072,64]
    const float*     W     = (const float*)d_in[3];      // [3,64,64]
    const float*     bias  = (const float*)d_in[4];      // [3,64]
    const float*     gamma = (const float*)d_in[5];      // [3,64]
    const float*     beta  = (const float*)d_in[6];      // [3,64]
    const float*     gemb  = (const float*)d_in[7];      // [4096,64]

    const size_t xElems = (size_t)NGRAPHS * NNODES * HD; // 2M floats = 8 MB
    float* xbuf = (float*)d_ws;                          // current features
    float* acc  = (float*)d_ws + xElems;                 // scatter accumulator
    float* hbuf = (float*)d_out;                         // reuse d_out as h scratch

    hipMemcpyAsync(xbuf, x0, xElems * sizeof(float), hipMemcpyDeviceToDevice, stream);

    const int totalRows = NGRAPHS * NNODES;              // 32768
    dim3 blk(256);

    for (int L = 0; L < NLAYERS; ++L) {
        gemm_wmma_f32<<<totalRows / 128, blk, 0, stream>>>(xbuf, W + L * HD * HD, hbuf);
        hipMemsetAsync(acc, 0, xElems * sizeof(float), stream);
        edge_scatter<<<(NGRAPHS * NEDGES) / 8, blk, 0, stream>>>(hbuf, ea, ei, acc);
        post_ln<<<totalRows / 8, blk, 0, stream>>>(acc, bias + L * HD, gamma + L * HD,
                                                   beta + L * HD, xbuf, L > 0);
    }
    finalize<<<totalRows / 8, blk, 0, stream>>>(x0, xbuf, gemb, (float*)d_out);
}